// DEC_GNN_Encoder_76948634075659
// MI455X (gfx1250) — compile-verified
//
#include <hip/hip_runtime.h>
#include <math.h>

// ---- problem constants (fixed by reference) ----
#define NNODES   27648
#define NEDGES   442368
#define FIN      256
#define HEADS    4
#define CPH      128          // per-head out channels
#define HC       512          // HEADS*CPH
#define HIDDEN   256
#define EDIM     16
#define NAG      48           // agents per group
#define NSEL     576          // 12*48 selected diagonal nodes
#define NSQ      2304         // 48*48
#define NP1      49           // n+1
#define MU_BLOCKS 432

typedef float v2f __attribute__((ext_vector_type(2)));
typedef float v8f __attribute__((ext_vector_type(8)));

__device__ __forceinline__ float lrelu(float v, float s) { return v > 0.f ? v : s * v; }
__device__ __forceinline__ int sel_id(int rank) { return (rank / NAG) * NSQ + (rank % NAG) * NP1; }

// ---------------- K0: zero atomics targets ----------------
__global__ void k_init(int* cnt, int* deg) {
    int t = blockIdx.x * blockDim.x + threadIdx.x;
    if (t == 0) *cnt = 0;
    if (t >= 32 && t < 32 + NSEL) deg[t - 32] = 0;
}

// ---------------- K1: blockwise partial sums of edge_attr (for mean) ----------------
__global__ void k_mu_part(const float* __restrict__ ea, float* __restrict__ mu_part) {
    __shared__ float sdata[256];
    int tid = threadIdx.x;
    int dim = tid & 15;
    int sub = tid >> 4;                 // 0..15 (edge within group of 16)
    float acc = 0.f;
    for (int e = blockIdx.x * 16 + sub; e < NEDGES; e += gridDim.x * 16)
        acc += ea[e * EDIM + dim];      // fully coalesced: base*16 + tid
    sdata[tid] = acc;
    __syncthreads();
    if (tid < 16) {
        float s = 0.f;
        for (int i = 0; i < 16; ++i) s += sdata[i * 16 + tid];
        mu_part[blockIdx.x * 16 + tid] = s;   // deterministic (no float atomics)
    }
}

// ---------------- K2: fold matrices P,Q,M and finalize mu ----------------
__global__ void k_fold(const float* __restrict__ W, const float* __restrict__ att_src,
                       const float* __restrict__ att_dst, const float* __restrict__ We,
                       const float* __restrict__ att_edge, const float* __restrict__ mu_part,
                       float* __restrict__ mu, float* __restrict__ P, float* __restrict__ Q,
                       float* __restrict__ M) {
    int g = blockIdx.x * blockDim.x + threadIdx.x;
    if (g < 1024) {                      // P[d*4+h] = sum_c W[d, h*128+c]*att_src[h,c]
        int d = g >> 2, h = g & 3;
        const float* wr = W + d * HC + h * CPH;
        const float* a  = att_src + h * CPH;
        float s = 0.f;
        for (int c = 0; c < CPH; ++c) s += wr[c] * a[c];
        P[g] = s;
    } else if (g < 2048) {               // Q (att_dst fold)
        int gg = g - 1024;
        int d = gg >> 2, h = gg & 3;
        const float* wr = W + d * HC + h * CPH;
        const float* a  = att_dst + h * CPH;
        float s = 0.f;
        for (int c = 0; c < CPH; ++c) s += wr[c] * a[c];
        Q[gg] = s;
    } else if (g < 2112) {               // M[d*4+h] from W_edge/att_edge
        int gg = g - 2048;
        int d = gg >> 2, h = gg & 3;
        const float* wr = We + d * HC + h * CPH;
        const float* a  = att_edge + h * CPH;
        float s = 0.f;
        for (int c = 0; c < CPH; ++c) s += wr[c] * a[c];
        M[gg] = s;
    } else if (g < 2128) {               // mu[d] = (1/E) * sum of partials (fixed order)
        int d = g - 2112;
        float s = 0.f;
        for (int b = 0; b < MU_BLOCKS; ++b) s += mu_part[b * 16 + d];
        mu[d] = s * (1.0f / (float)NEDGES);
    }
}

// ---------------- K3: a_src / a_dst at the 576 selected nodes ----------------
__global__ void k_sel_att(const float* __restrict__ x, const float* __restrict__ P,
                          const float* __restrict__ Q, float* __restrict__ a_src_sel,
                          float* __restrict__ a_dst_sel) {
    int rank = blockIdx.x;
    int id = sel_id(rank);
    __shared__ float xs[FIN];
    int tid = threadIdx.x;               // 128 threads
    xs[tid]       = x[(size_t)id * FIN + tid];
    xs[tid + 128] = x[(size_t)id * FIN + tid + 128];
    __syncthreads();
    int w = tid >> 5, lane = tid & 31;   // warp w handles head w
    float sp = 0.f, sq = 0.f;
    for (int d = lane; d < FIN; d += 32) {
        float v = xs[d];
        sp += v * P[d * 4 + w];
        sq += v * Q[d * 4 + w];
    }
    for (int off = 16; off > 0; off >>= 1) {
        sp += __shfl_down(sp, off);
        sq += __shfl_down(sq, off);
    }
    if (lane == 0) {
        a_src_sel[rank * 4 + w] = sp;
        a_dst_sel[rank * 4 + w] = sq;
    }
}

// ---------------- K5: filter edges with selected dst; compact + per-edge base logits ----------------
__global__ void k_filter(const int* __restrict__ ei, const float* __restrict__ x,
                         const float* __restrict__ ea, const float* __restrict__ P,
                         const float* __restrict__ M, int* __restrict__ cnt,
                         int* __restrict__ deg, int* __restrict__ comp_src,
                         int* __restrict__ comp_rank, float* __restrict__ comp_base) {
    int e = blockIdx.x * blockDim.x + threadIdx.x;
    if (e >= NEDGES) return;
    int dst = ei[NEDGES + e];
    int r = dst % NSQ;
    if (r % NP1 != 0) return;            // dst not on the diagonal: dead edge
    int rank = (dst / NSQ) * NAG + r / NP1;
    int src = ei[e];
    float b0 = 0.f, b1 = 0.f, b2 = 0.f, b3 = 0.f;
    const float* xr = x + (size_t)src * FIN;
    for (int d = 0; d < FIN; ++d) {      // a_src[src] = x[src] @ P
        float v = xr[d];
        b0 += v * P[d * 4 + 0]; b1 += v * P[d * 4 + 1];
        b2 += v * P[d * 4 + 2]; b3 += v * P[d * 4 + 3];
    }
    const float* er = ea + (size_t)e * EDIM;
    for (int d = 0; d < EDIM; ++d) {     // a_edge = ea[e] @ M
        float v = er[d];
        b0 += v * M[d * 4 + 0]; b1 += v * M[d * 4 + 1];
        b2 += v * M[d * 4 + 2]; b3 += v * M[d * 4 + 3];
    }
    int idx = atomicAdd(cnt, 1);
    comp_src[idx]  = src;
    comp_rank[idx] = rank;
    comp_base[idx * 4 + 0] = b0; comp_base[idx * 4 + 1] = b1;
    comp_base[idx * 4 + 2] = b2; comp_base[idx * 4 + 3] = b3;
    atomicAdd(&deg[rank], 1);
}

// ---------------- K6: tiny exclusive scan over 576 degrees ----------------
__global__ void k_scan(const int* __restrict__ deg, int* __restrict__ offs,
                       int* __restrict__ cursor) {
    if (blockIdx.x == 0 && threadIdx.x == 0) {
        int acc = 0;
        for (int i = 0; i < NSEL; ++i) { offs[i] = acc; cursor[i] = acc; acc += deg[i]; }
        offs[NSEL] = acc;
    }
}

// ---------------- K7: scatter compacted edges into CSR buckets ----------------
__global__ void k_scatter(const int* __restrict__ cnt, const int* __restrict__ comp_rank,
                          int* __restrict__ cursor, int* __restrict__ edge_order) {
    int n = *cnt;
    for (int i = blockIdx.x * blockDim.x + threadIdx.x; i < n; i += gridDim.x * blockDim.x) {
        int pos = atomicAdd(&cursor[comp_rank[i]], 1);
        edge_order[pos] = i;
    }
}

// ---------------- K8: per-dst segment softmax + alpha-weighted x accumulation ----------------
__global__ void k_aggregate(const float* __restrict__ x, const float* __restrict__ mu,
                            const float* __restrict__ M, const float* __restrict__ a_dst_sel,
                            const float* __restrict__ a_src_sel, const int* __restrict__ offs,
                            const int* __restrict__ edge_order, const int* __restrict__ comp_src,
                            const float* __restrict__ comp_base, float* __restrict__ Y) {
    __shared__ float s_red[256];
    __shared__ float s_adst[4], s_logit_self[4], s_m[4], s_den[4];
    __shared__ float s_alpha[64 * 4];
    __shared__ int   s_srcs[64];
    int rank = blockIdx.x;
    int tid = threadIdx.x;               // 256 threads
    int id = sel_id(rank);
    int e0 = offs[rank], e1 = offs[rank + 1];
    int deg = e1 - e0;

    if (tid < 4) {
        float al = 0.f;
        for (int d = 0; d < EDIM; ++d) al += mu[d] * M[d * 4 + tid];   // self-loop a_edge
        float adst  = a_dst_sel[rank * 4 + tid];
        float asrc  = a_src_sel[rank * 4 + tid];
        s_adst[tid] = adst;
        s_logit_self[tid] = lrelu(asrc + adst + al, 0.2f);
    }
    __syncthreads();

    // phase 1a: per-head max over {edges, self}
    for (int h = 0; h < 4; ++h) {
        float adst = s_adst[h];
        float lm = s_logit_self[h];
        for (int j = tid; j < deg; j += 256) {
            int eidx = edge_order[e0 + j];
            lm = fmaxf(lm, lrelu(comp_base[eidx * 4 + h] + adst, 0.2f));
        }
        s_red[tid] = lm; __syncthreads();
        for (int s = 128; s > 0; s >>= 1) {
            if (tid < s) s_red[tid] = fmaxf(s_red[tid], s_red[tid + s]);
            __syncthreads();
        }
        if (tid == 0) s_m[h] = s_red[0];
        __syncthreads();
    }
    // phase 1b: per-head sum of exp
    for (int h = 0; h < 4; ++h) {
        float adst = s_adst[h], m = s_m[h];
        float ls = (tid == 0) ? expf(s_logit_self[h] - m) : 0.f;
        for (int j = tid; j < deg; j += 256) {
            int eidx = edge_order[e0 + j];
            ls += expf(lrelu(comp_base[eidx * 4 + h] + adst, 0.2f) - m);
        }
        s_red[tid] = ls; __syncthreads();
        for (int s = 128; s > 0; s >>= 1) {
            if (tid < s) s_red[tid] += s_red[tid + s];
            __syncthreads();
        }
        if (tid == 0) s_den[h] = s_red[0] + 1e-16f;
        __syncthreads();
    }

    // phase 2: Y[rank,h,col] = sum_e alpha[e,h]*x[src_e,col] + alpha_self[h]*x[id,col]
    float acc[4] = {0.f, 0.f, 0.f, 0.f};
    int col = tid;                        // one of 256 input features
    for (int base = 0; base < deg; base += 64) {
        int nch = deg - base; if (nch > 64) nch = 64;
        if (tid < nch) {
            int eidx = edge_order[e0 + base + tid];
            s_srcs[tid] = comp_src[eidx];
            for (int h = 0; h < 4; ++h) {
                float l = lrelu(comp_base[eidx * 4 + h] + s_adst[h], 0.2f);
                s_alpha[tid * 4 + h] = expf(l - s_m[h]) / s_den[h];
            }
        }
        __syncthreads();
        for (int j = 0; j < nch; ++j) {
            float v = x[(size_t)s_srcs[j] * FIN + col];
            acc[0] += s_alpha[j * 4 + 0] * v;
            acc[1] += s_alpha[j * 4 + 1] * v;
            acc[2] += s_alpha[j * 4 + 2] * v;
            acc[3] += s_alpha[j * 4 + 3] * v;
        }
        __syncthreads();
    }
    float vs = x[(size_t)id * FIN + col];
    for (int h = 0; h < 4; ++h) {
        float as = expf(s_logit_self[h] - s_m[h]) / s_den[h];
        acc[h] += as * vs;
        Y[(size_t)rank * (HEADS * FIN) + h * FIN + col] = acc[h];
    }
}

// ---------------- K9: per-head WMMA GEMM  h2 = lrelu(Y_h @ W_h + bias, 0.01) ----------------
// One wave per 16x16 output tile; K stepped by 4 with V_WMMA_F32_16X16X4_F32.
__global__ void k_gemm_gat(const float* __restrict__ Y, const float* __restrict__ W,
                           const float* __restrict__ bias, float* __restrict__ h2) {
    int wave = (blockIdx.x * blockDim.x + threadIdx.x) >> 5;
    int lane = threadIdx.x & 31;
    int row_tile = wave >> 5;            // 36 row tiles
    int col_tile = wave & 31;            // 32 col tiles over HC=512
    if (row_tile >= NSEL / 16) return;   // wave-uniform
    int row0 = row_tile * 16;
    int col0 = col_tile * 16;
    int head = col_tile >> 3;            // 8 col tiles per head
    const float* Yh = Y + head * FIN;    // Y[row][head][k] = Y[row*1024 + head*256 + k]
    int arow = row0 + (lane & 15);
    int koff = (lane < 16) ? 0 : 2;      // A: lanes 0-15 hold K,K+1 ; lanes 16-31 hold K+2,K+3
    int bcol = col0 + (lane & 15);
    int boff = (lane < 16) ? 0 : 1;      // B: lanes 0-15 row K ; lanes 16-31 row K+1 (v1: +2)
    v8f c = {0.f, 0.f, 0.f, 0.f, 0.f, 0.f, 0.f, 0.f};
    for (int k = 0; k < FIN; k += 4) {
        v2f a, b;
        a.x = Yh[(size_t)arow * (HEADS * FIN) + k + koff];
        a.y = Yh[(size_t)arow * (HEADS * FIN) + k + koff + 1];
        b.x = W[(size_t)(k + boff) * HC + bcol];
        b.y = W[(size_t)(k + boff + 2) * HC + bcol];
        c = __builtin_amdgcn_wmma_f32_16x16x4_f32(false, a, false, b, (short)0, c,
                                                  false, false);
    }
    int crow0 = row0 + ((lane < 16) ? 0 : 8);
    int ccol = col0 + (lane & 15);
    float bi = bias[ccol];
    for (int i = 0; i < 8; ++i) {
        float v = lrelu(c[i] + bi, 0.01f);
        h2[(size_t)(crow0 + i) * HC + ccol] = v;
    }
}

// ---------------- K10: WMMA GEMM  out = lrelu(h2 @ fc_W + fc_b, 0.01) ----------------
__global__ void k_gemm_fc(const float* __restrict__ h2, const float* __restrict__ fcW,
                          const float* __restrict__ fcb, float* __restrict__ out) {
    int wave = (blockIdx.x * blockDim.x + threadIdx.x) >> 5;
    int lane = threadIdx.x & 31;
    int row_tile = wave >> 4;            // 36 row tiles
    int col_tile = wave & 15;            // 16 col tiles over HIDDEN=256
    if (row_tile >= NSEL / 16) return;
    int row0 = row_tile * 16;
    int col0 = col_tile * 16;
    int arow = row0 + (lane & 15);
    int koff = (lane < 16) ? 0 : 2;
    int bcol = col0 + (lane & 15);
    int boff = (lane < 16) ? 0 : 1;
    v8f c = {0.f, 0.f, 0.f, 0.f, 0.f, 0.f, 0.f, 0.f};
    for (int k = 0; k < HC; k += 4) {
        v2f a, b;
        a.x = h2[(size_t)arow * HC + k + koff];
        a.y = h2[(size_t)arow * HC + k + koff + 1];
        b.x = fcW[(size_t)(k + boff) * HIDDEN + bcol];
        b.y = fcW[(size_t)(k + boff + 2) * HIDDEN + bcol];
        c = __builtin_amdgcn_wmma_f32_16x16x4_f32(false, a, false, b, (short)0, c,
                                                  false, false);
    }
    int crow0 = row0 + ((lane < 16) ? 0 : 8);
    int ccol = col0 + (lane & 15);
    float bi = fcb[ccol];
    for (int i = 0; i < 8; ++i) {
        float v = lrelu(c[i] + bi, 0.01f);
        out[(size_t)(crow0 + i) * HIDDEN + ccol] = v;
    }
}

// ---------------- launcher ----------------
extern "C" void kernel_launch(void* const* d_in, const int* in_sizes, int n_in,
                              void* d_out, int out_size, void* d_ws, size_t ws_size,
                              hipStream_t stream) {
    (void)in_sizes; (void)n_in; (void)out_size; (void)ws_size;
    const float* x        = (const float*)d_in[0];
    const int*   ei       = (const int*)d_in[1];
    const float* ea       = (const float*)d_in[2];
    // d_in[3] num_groups, d_in[4] agents_per_group: fixed (12, 48)
    const float* W        = (const float*)d_in[5];
    const float* att_src  = (const float*)d_in[6];
    const float* att_dst  = (const float*)d_in[7];
    const float* We       = (const float*)d_in[8];
    const float* att_edge = (const float*)d_in[9];
    const float* bias     = (const float*)d_in[10];
    const float* fcW      = (const float*)d_in[11];
    const float* fcb      = (const float*)d_in[12];
    float* out = (float*)d_out;

    // workspace carving (4-byte elements)
    float* mu_part   = (float*)d_ws;                       // 432*16
    float* mu        = mu_part + MU_BLOCKS * 16;           // 16
    float* P         = mu + 16;                            // 1024
    float* Q         = P + 1024;                           // 1024
    float* M         = Q + 1024;                           // 64
    float* a_dst_sel = M + 64;                             // 2304
    float* a_src_sel = a_dst_sel + NSEL * HEADS;           // 2304
    int*   cnt       = (int*)(a_src_sel + NSEL * HEADS);   // 1
    int*   deg       = cnt + 1;                            // 576
    int*   offs      = deg + NSEL;                         // 577
    int*   cursor    = offs + NSEL + 1;                    // 576
    int*   comp_src  = cursor + NSEL;                      // E
    int*   comp_rank = comp_src + NEDGES;                  // E
    float* comp_base = (float*)(comp_rank + NEDGES);       // 4E
    int*   edge_order= (int*)(comp_base + 4 * (size_t)NEDGES); // E
    float* Yb        = (float*)(edge_order + NEDGES);      // 576*4*256
    float* h2        = Yb + (size_t)NSEL * HEADS * FIN;    // 576*512

    k_init     <<<1, 640, 0, stream>>>(cnt, deg);
    k_mu_part  <<<MU_BLOCKS, 256, 0, stream>>>(ea, mu_part);
    k_fold     <<<9, 256, 0, stream>>>(W, att_src, att_dst, We, att_edge, mu_part,
                                       mu, P, Q, M);
    k_sel_att  <<<NSEL, 128, 0, stream>>>(x, P, Q, a_src_sel, a_dst_sel);
    k_filter   <<<(NEDGES + 255) / 256, 256, 0, stream>>>(ei, x, ea, P, M, cnt, deg,
                                                          comp_src, comp_rank, comp_base);
    k_scan     <<<1, 1, 0, stream>>>(deg, offs, cursor);
    k_scatter  <<<1024, 256, 0, stream>>>(cnt, comp_rank, cursor, edge_order);
    k_aggregate<<<NSEL, 256, 0, stream>>>(x, mu, M, a_dst_sel, a_src_sel, offs,
                                          edge_order, comp_src, comp_base, Yb);
    k_gemm_gat <<<(NSEL / 16) * 32 / 4, 128, 0, stream>>>(Yb, W, bias, h2);   // 288 blocks
    k_gemm_fc  <<<(NSEL / 16) * 16 / 4, 128, 0, stream>>>(h2, fcW, fcb, out); // 144 blocks
}